// GraphConv_3092376453557
// MI455X (gfx1250) — compile-verified
//
#include <hip/hip_runtime.h>

typedef __attribute__((ext_vector_type(2))) float v2f;
typedef __attribute__((ext_vector_type(8))) float v8f;

#define IN_DIM  256
#define OUT_DIM 256

// ---------------------------------------------------------------------------
// Kernel 0: zero the aggregation buffer (float4 stores, grid-stride free)
// ---------------------------------------------------------------------------
__global__ void gc_zero_kernel(float4* __restrict__ p, long long n4) {
    long long i = (long long)blockIdx.x * blockDim.x + threadIdx.x;
    if (i < n4) p[i] = make_float4(0.f, 0.f, 0.f, 0.f);
}

// ---------------------------------------------------------------------------
// Kernel 1: edge scatter.  One wave32 per edge.
//   lanes cover 256 floats as 2x float4 per lane (coalesced gather from L2-
//   resident x), then 8 non-returning global_atomic_add_f32 per lane.
// ---------------------------------------------------------------------------
__global__ void gc_scatter_kernel(const float* __restrict__ x,
                                  const int*   __restrict__ erow,
                                  const int*   __restrict__ ecol,
                                  const float* __restrict__ eval,
                                  float*       __restrict__ agg,
                                  int n_edges) {
    int wave = (int)((blockIdx.x * (long long)blockDim.x + threadIdx.x) >> 5);
    int lane = threadIdx.x & 31;
    if (wave >= n_edges) return;

    int   r = erow[wave];            // uniform per wave -> scalar loads
    int   c = ecol[wave];
    float v = eval[wave];

    const float4* src = (const float4*)(x + (long long)c * IN_DIM);
    float*        dst = agg + (long long)r * IN_DIM;

#pragma unroll
    for (int h = 0; h < 2; ++h) {
        int    q = lane + 32 * h;          // float4 index 0..63
        float4 f = src[q];
        int    b = q * 4;
        atomicAdd(dst + b + 0, v * f.x);
        atomicAdd(dst + b + 1, v * f.y);
        atomicAdd(dst + b + 2, v * f.z);
        atomicAdd(dst + b + 3, v * f.w);
    }
}

// ---------------------------------------------------------------------------
// Kernel 2: out = relu(agg @ W^T) with V_WMMA_F32_16X16X4_F32.
//   One wave per 16x16 output tile.  M = n_nodes (multiple of 16),
//   N = OUT_DIM, K = IN_DIM.  A = agg (MxK row-major), B[k][n] = W[n][k]
//   (W is [out,in] row-major -> contiguous float2 per fragment, no transpose).
//
//   A frag (16x4 f32): lane L holds m = L&15; half = L>>4 selects K pair
//     VGPR0 = A[m][k0+2*half], VGPR1 = A[m][k0+2*half+1]     -> float2 load
//   B frag (4x16 f32): lane L holds n = L&15; same K-pair striping -> float2
//   D (16x16 f32, 8 VGPRs): VGPR j -> row (j + 8*half), col = L&15
// ---------------------------------------------------------------------------
__global__ void gc_gemm_relu_kernel(const float* __restrict__ agg,
                                    const float* __restrict__ W,
                                    float*       __restrict__ out,
                                    int n_nodes) {
    const int ntilesN = OUT_DIM / 16;  // 16
    int waveId = (int)((blockIdx.x * (long long)blockDim.x + threadIdx.x) >> 5);
    int lane   = threadIdx.x & 31;

    int mtile = waveId / ntilesN;
    int ntile = waveId % ntilesN;
    if (mtile * 16 >= n_nodes) return;          // uniform per wave: EXEC stays all-ones

    int half = lane >> 4;                        // 0: K pair {0,1}, 1: K pair {2,3}
    int mn   = lane & 15;

    const float* arow = agg + (long long)(mtile * 16 + mn) * IN_DIM + 2 * half;
    const float* brow = W   + (long long)(ntile * 16 + mn) * IN_DIM + 2 * half;

    v8f c = {};
#pragma unroll 8
    for (int k0 = 0; k0 < IN_DIM; k0 += 4) {
        v2f a = *(const v2f*)(arow + k0);
        v2f b = *(const v2f*)(brow + k0);
        c = __builtin_amdgcn_wmma_f32_16x16x4_f32(
                /*neg_a=*/false, a, /*neg_b=*/false, b,
                /*c_mod=*/(short)0, c, /*reuse_a=*/false, /*reuse_b=*/false);
    }

    // fused ReLU epilogue
    int ncol  = ntile * 16 + mn;
    int mbase = mtile * 16 + 8 * half;
#pragma unroll
    for (int j = 0; j < 8; ++j) {
        float vv = c[j];
        vv = vv > 0.f ? vv : 0.f;
        out[(long long)(mbase + j) * OUT_DIM + ncol] = vv;
    }
}

// ---------------------------------------------------------------------------
// Launch: inputs are (x, edge_row, edge_col, edge_val, W) per setup_inputs().
// d_ws holds agg [n_nodes * IN_DIM] f32.
// ---------------------------------------------------------------------------
extern "C" void kernel_launch(void* const* d_in, const int* in_sizes, int n_in,
                              void* d_out, int out_size, void* d_ws, size_t ws_size,
                              hipStream_t stream) {
    const float* x    = (const float*)d_in[0];
    const int*   erow = (const int*)  d_in[1];
    const int*   ecol = (const int*)  d_in[2];
    const float* ev   = (const float*)d_in[3];
    const float* W    = (const float*)d_in[4];
    float*       out  = (float*)d_out;
    float*       agg  = (float*)d_ws;

    int n_nodes = in_sizes[0] / IN_DIM;   // 50000
    int n_edges = in_sizes[1];            // 1,600,000

    // 1) agg = 0
    long long n4 = (long long)n_nodes * IN_DIM / 4;
    int zgrid = (int)((n4 + 255) / 256);
    gc_zero_kernel<<<zgrid, 256, 0, stream>>>((float4*)agg, n4);

    // 2) scatter: one wave32 per edge, 8 waves per 256-thread block
    int sgrid = (n_edges + 7) / 8;
    gc_scatter_kernel<<<sgrid, 256, 0, stream>>>(x, erow, ecol, ev, agg, n_edges);

    // 3) WMMA GEMM + ReLU: one wave per 16x16 tile, 8 waves per block
    int ntiles = (n_nodes / 16) * (OUT_DIM / 16);
    int ggrid  = (ntiles + 7) / 8;
    gc_gemm_relu_kernel<<<ggrid, 256, 0, stream>>>(agg, W, out, n_nodes);
}